// GATModel_65515431133470
// MI455X (gfx1250) — compile-verified
//
#include <hip/hip_runtime.h>
#include <math.h>

// ---------------------------------------------------------------------------
// 2-layer GAT (PyG GATConv semantics) for MI455X / gfx1250.
// Dense 16-K feature transforms use V_WMMA_F32_16X16X4_F32 (fp32-exact).
// Edge softmax is fused: atomic-max pass + single weighted-accumulate pass
// (sum w, sum w*h), normalized in a node pass.
// ---------------------------------------------------------------------------

typedef __attribute__((ext_vector_type(2))) float v2f;
typedef __attribute__((ext_vector_type(8))) float v8f;

// monotone float <-> uint mapping for atomic max on floats
__device__ __forceinline__ unsigned encf(float f) {
  unsigned u = __float_as_uint(f);
  return (u & 0x80000000u) ? ~u : (u | 0x80000000u);
}
__device__ __forceinline__ float decf(unsigned u) {
  unsigned v = (u & 0x80000000u) ? (u & 0x7FFFFFFFu) : ~u;
  return __uint_as_float(v);
}
#define ENC_NEG_INF 0x007FFFFFu   // encf(-inf)

// ---------------------------------------------------------------------------
// GEMM: Hout[nrows x ncols] = X[nrows x 16] * W[16 x ncols], ncols <= 16.
// One wave per 16-row tile, K=16 via 4x V_WMMA_F32_16X16X4_F32.
// Branch-free: out-of-range rows/cols are CLAMPED on load (they only feed
// D elements that are never stored), so no exec-mask regions around loads.
// ---------------------------------------------------------------------------
__global__ void gemm16_wmma(const float* __restrict__ X,
                            const float* __restrict__ W,
                            float* __restrict__ Hout,
                            int nrows, int ncols) {
  const int wave = blockIdx.x * (blockDim.x >> 5) + (threadIdx.x >> 5);
  const int lane = threadIdx.x & 31;
  const int row_base = wave * 16;
  if (row_base >= nrows) return;                 // wave-uniform exit

  const int m    = lane & 15;                    // A row / B,D column in tile
  const int half = lane >> 4;                    // 0 or 1 (K pair select)
  int row = row_base + m; if (row > nrows - 1) row = nrows - 1;   // clamp
  int col = m;            if (col > ncols - 1) col = ncols - 1;   // clamp

  const float2* __restrict__ Xr = (const float2*)(X + (size_t)row * 16);

  v8f c = {0.f, 0.f, 0.f, 0.f, 0.f, 0.f, 0.f, 0.f};
#pragma unroll
  for (int k0 = 0; k0 < 4; ++k0) {
    const int k = k0 * 4 + half * 2;
    // A 16x4 layout: lanes 0-15 -> (M=lane, K={k,k+1}); lanes 16-31 -> K+2
    const float2 av = Xr[k0 * 2 + half];         // one b64 load
    v2f a; a.x = av.x; a.y = av.y;
    // B 4x16 layout (mirrored): lane gives column, half gives K pair
    v2f b;
    b.x = W[(k)     * ncols + col];
    b.y = W[(k + 1) * ncols + col];
    c = __builtin_amdgcn_wmma_f32_16x16x4_f32(
        /*neg_a=*/false, a, /*neg_b=*/false, b,
        /*c_mod=*/(short)0, c, /*reuse_a=*/false, /*reuse_b=*/false);
  }

  // D layout: VGPR r holds M=r (lanes 0-15) / M=r+8 (lanes 16-31), N=lane&15
  if (m < ncols) {
    const int rofs = half * 8;
    float* __restrict__ outp = Hout + (size_t)(row_base + rofs) * ncols + m;
    if (row_base + 16 <= nrows) {                // full tile: unconditional
#pragma unroll
      for (int r = 0; r < 8; ++r) outp[(size_t)r * ncols] = c[r];
    } else {
#pragma unroll
      for (int r = 0; r < 8; ++r)
        if (row_base + rofs + r < nrows) outp[(size_t)r * ncols] = c[r];
    }
  }
}

// ---------------------------------------------------------------------------
// Per-node: attention logits al_s/al_d (H=2 heads), init max/z/acc buffers.
// ---------------------------------------------------------------------------
template <int C>
__global__ void node_logits_init(const float* __restrict__ Hm,
                                 const float* __restrict__ a_src,
                                 const float* __restrict__ a_dst,
                                 float2* __restrict__ als, float2* __restrict__ ald,
                                 unsigned* __restrict__ menc,
                                 float2* __restrict__ z, float* __restrict__ acc,
                                 int nrows) {
  const int n = blockIdx.x * blockDim.x + threadIdx.x;
  if (n >= nrows) return;
  constexpr int F = 2 * C;
  const float4* __restrict__ hr = (const float4*)(Hm + (size_t)n * F);
  float s0 = 0.f, s1 = 0.f, d0 = 0.f, d1 = 0.f;
#pragma unroll
  for (int v = 0; v < C / 4; ++v) {
    const float4 h0 = hr[v];                // head 0 channels 4v..4v+3
    const float4 h1 = hr[C / 4 + v];        // head 1 channels 4v..4v+3
#pragma unroll
    for (int j = 0; j < 4; ++j) {
      const float e0 = (&h0.x)[j], e1 = (&h1.x)[j];
      const int cch = 4 * v + j;
      s0 += e0 * a_src[cch];     d0 += e0 * a_dst[cch];
      s1 += e1 * a_src[C + cch]; d1 += e1 * a_dst[C + cch];
    }
  }
  als[n] = make_float2(s0, s1);
  ald[n] = make_float2(d0, d1);
  menc[n * 2] = ENC_NEG_INF; menc[n * 2 + 1] = ENC_NEG_INF;
  z[n] = make_float2(0.f, 0.f);
  float4* __restrict__ ar = (float4*)(acc + (size_t)n * F);
#pragma unroll
  for (int v = 0; v < F / 4; ++v) ar[v] = make_float4(0.f, 0.f, 0.f, 0.f);
}

// ---------------------------------------------------------------------------
// Edge pass 1: per-dst segment max of leaky_relu(al_s[src]+al_d[dst]).
// Threads [0,E) are real edges, [E, E+N) are the implicit self-loops.
// ---------------------------------------------------------------------------
__global__ void edge_max(const int* __restrict__ ei,
                         const float2* __restrict__ als,
                         const float2* __restrict__ ald,
                         unsigned* __restrict__ menc, int E, int N) {
  const int t = blockIdx.x * blockDim.x + threadIdx.x;
  if (t >= E + N) return;
  int src, dst;
  if (t < E) { src = ei[t]; dst = ei[E + t]; } else { src = dst = t - E; }
  const float2 s = als[src];
  const float2 d = ald[dst];
  float e0 = s.x + d.x; e0 = (e0 > 0.f) ? e0 : 0.2f * e0;
  float e1 = s.y + d.y; e1 = (e1 > 0.f) ? e1 : 0.2f * e1;
  atomicMax(&menc[dst * 2],     encf(e0));
  atomicMax(&menc[dst * 2 + 1], encf(e1));
}

// ---------------------------------------------------------------------------
// Edge pass 2: w = exp(e - m[dst]); z[dst] += w; acc[dst] += w * h[src].
// ---------------------------------------------------------------------------
template <int C>
__global__ void edge_accum(const int* __restrict__ ei,
                           const float2* __restrict__ als,
                           const float2* __restrict__ ald,
                           const unsigned* __restrict__ menc,
                           float* __restrict__ z, float* __restrict__ acc,
                           const float* __restrict__ Hm,
                           int E, int N) {
  const int t = blockIdx.x * blockDim.x + threadIdx.x;
  if (t >= E + N) return;
  int src, dst;
  if (t < E) { src = ei[t]; dst = ei[E + t]; } else { src = dst = t - E; }
  constexpr int F = 2 * C;
  const float2 s = als[src];
  const float2 d = ald[dst];
  const float4* __restrict__ hs = (const float4*)(Hm + (size_t)src * F);
  float w[2];
#pragma unroll
  for (int hd = 0; hd < 2; ++hd) {
    float e = (hd ? s.y + d.y : s.x + d.x);
    e = (e > 0.f) ? e : 0.2f * e;
    float mm = decf(menc[dst * 2 + hd]);
    if (!(mm >= -3.0e38f && mm <= 3.0e38f)) mm = 0.f;  // isfinite guard
    w[hd] = __expf(e - mm);
    atomicAdd(&z[dst * 2 + hd], w[hd]);
  }
  float* __restrict__ ad = acc + (size_t)dst * F;
#pragma unroll
  for (int hd = 0; hd < 2; ++hd) {
#pragma unroll
    for (int v = 0; v < C / 4; ++v) {
      const float4 hv = hs[hd * (C / 4) + v];
      const int base = hd * C + 4 * v;
      atomicAdd(&ad[base + 0], w[hd] * hv.x);
      atomicAdd(&ad[base + 1], w[hd] * hv.y);
      atomicAdd(&ad[base + 2], w[hd] * hv.z);
      atomicAdd(&ad[base + 3], w[hd] * hv.w);
    }
  }
}

// ---------------------------------------------------------------------------
// Per-node finalize: out = acc / (z + 1e-16) + bias, optional ELU.
// ---------------------------------------------------------------------------
template <int C, bool ELU>
__global__ void node_finalize(const float* __restrict__ acc,
                              const float2* __restrict__ z,
                              const float* __restrict__ bias,
                              float* __restrict__ out, int nrows) {
  const int n = blockIdx.x * blockDim.x + threadIdx.x;
  if (n >= nrows) return;
  constexpr int F = 2 * C;
  const float4* __restrict__ ar = (const float4*)(acc + (size_t)n * F);
  float4* __restrict__ outp = (float4*)(out + (size_t)n * F);
  const float2 zz = z[n];
  const float rz0 = 1.f / (zz.x + 1e-16f);
  const float rz1 = 1.f / (zz.y + 1e-16f);
#pragma unroll
  for (int v = 0; v < F / 4; ++v) {
    float4 a = ar[v];
    const float rz = (4 * v < C) ? rz0 : rz1;   // C multiple of 4 -> uniform
#pragma unroll
    for (int j = 0; j < 4; ++j) {
      float x = (&a.x)[j] * rz + bias[4 * v + j];
      if (ELU) x = (x > 0.f) ? x : (__expf(x) - 1.f);
      (&a.x)[j] = x;
    }
    outp[v] = a;
  }
}

// ---------------------------------------------------------------------------
extern "C" void kernel_launch(void* const* d_in, const int* in_sizes, int n_in,
                              void* d_out, int out_size, void* d_ws, size_t ws_size,
                              hipStream_t stream) {
  const float* x   = (const float*)d_in[0];
  const int*   ei  = (const int*)  d_in[1];
  const float* W1  = (const float*)d_in[2];
  const float* as1 = (const float*)d_in[3];
  const float* ad1 = (const float*)d_in[4];
  const float* b1  = (const float*)d_in[5];
  const float* W2  = (const float*)d_in[6];
  const float* as2 = (const float*)d_in[7];
  const float* ad2 = (const float*)d_in[8];
  const float* b2  = (const float*)d_in[9];

  const int N = in_sizes[0] / 16;   // 100000
  const int E = in_sizes[1] / 2;    // 3200000

  // workspace layout (floats): total 56*N = ~22.4 MB
  float*    ws   = (float*)d_ws;
  float*    h1   = ws;                               // 16N layer-1 features
  float*    acc1 = ws + (size_t)16 * N;              // 16N accum -> ELU'd x2
  float2*   als  = (float2*)(ws + (size_t)32 * N);   // 2N (reused)
  float2*   ald  = (float2*)(ws + (size_t)34 * N);   // 2N
  unsigned* menc = (unsigned*)(ws + (size_t)36 * N); // 2N
  float*    z    = ws + (size_t)38 * N;              // 2N
  float*    h2   = ws + (size_t)40 * N;              // 8N layer-2 features
  float*    acc2 = ws + (size_t)48 * N;              // 8N

  const int TB = 256;
  const int node_grid = (N + TB - 1) / TB;
  const int edge_grid = (E + N + TB - 1) / TB;
  const int tiles     = (N + 15) / 16;
  const int gemm_grid = (tiles + 7) / 8;             // 8 waves / block

  // ---- layer 1: 16 -> 2x8 ----
  gemm16_wmma<<<gemm_grid, TB, 0, stream>>>(x, W1, h1, N, 16);
  node_logits_init<8><<<node_grid, TB, 0, stream>>>(h1, as1, ad1, als, ald, menc,
                                                    (float2*)z, acc1, N);
  edge_max<<<edge_grid, TB, 0, stream>>>(ei, als, ald, menc, E, N);
  edge_accum<8><<<edge_grid, TB, 0, stream>>>(ei, als, ald, menc, z, acc1, h1, E, N);
  node_finalize<8, true><<<node_grid, TB, 0, stream>>>(acc1, (const float2*)z, b1, acc1, N);

  // ---- layer 2: 16 -> 2x4 ----
  gemm16_wmma<<<gemm_grid, TB, 0, stream>>>(acc1, W2, h2, N, 8);
  node_logits_init<4><<<node_grid, TB, 0, stream>>>(h2, as2, ad2, als, ald, menc,
                                                    (float2*)z, acc2, N);
  edge_max<<<edge_grid, TB, 0, stream>>>(ei, als, ald, menc, E, N);
  edge_accum<4><<<edge_grid, TB, 0, stream>>>(ei, als, ald, menc, z, acc2, h2, E, N);
  node_finalize<4, false><<<node_grid, TB, 0, stream>>>(acc2, (const float2*)z, b2,
                                                        (float*)d_out, N);
}